// BatchHardContrastiveLoss_47502338294213
// MI455X (gfx1250) — compile-verified
//
#include <hip/hip_runtime.h>
#include <hip/hip_bf16.h>

typedef __attribute__((ext_vector_type(2))) float v2f;
typedef __attribute__((ext_vector_type(8))) float v8f;

#define POS_THRESH 0.1f
#define NEG_THRESH 1.4f
#define MIN_DIST2  0.0225f   // 0.15^2
#define BIG_VAL    1.0e12f
#define WPB        8         // waves per block (256 threads, wave32)

static __device__ __forceinline__ v8f wmma_f32_16x16x4(v2f a, v2f b, v8f c) {
  // D = A(16x4 f32) x B(4x16 f32) + C(16x16 f32)
  return __builtin_amdgcn_wmma_f32_16x16x4_f32(
      /*neg_a=*/false, a, /*neg_b=*/false, b,
      /*c_mod=*/(short)0, c, /*reuse_a=*/false, /*reuse_b=*/false);
}

// ---------------------------------------------------------------------------
// Phase 1: gather pairs, compute norms, xyz (padded to 4), per-row pos term.
// ---------------------------------------------------------------------------
__global__ __launch_bounds__(256) void gather_kernel(
    const float* __restrict__ F0, const float* __restrict__ F1,
    const int* __restrict__ pairs, const float* __restrict__ xyz,
    float* __restrict__ pf0, float* __restrict__ pf1, float* __restrict__ sx,
    float* __restrict__ na, float* __restrict__ nb, float* __restrict__ nx,
    float* __restrict__ posr, int P) {
  int i = blockIdx.x * blockDim.x + threadIdx.x;
  if (i >= P) return;
  int i0 = pairs[2 * i];
  int i1 = pairs[2 * i + 1];
  const float4* r0 = reinterpret_cast<const float4*>(F0 + (size_t)i0 * 32);
  const float4* r1 = reinterpret_cast<const float4*>(F1 + (size_t)i1 * 32);
  float4* w0 = reinterpret_cast<float4*>(pf0 + (size_t)i * 32);
  float4* w1 = reinterpret_cast<float4*>(pf1 + (size_t)i * 32);
  float s0 = 0.f, s1 = 0.f, fp = 0.f;
#pragma unroll
  for (int d = 0; d < 8; ++d) {
    float4 x0 = r0[d];
    float4 x1 = r1[d];
    w0[d] = x0;
    w1[d] = x1;
    s0 += x0.x * x0.x + x0.y * x0.y + x0.z * x0.z + x0.w * x0.w;
    s1 += x1.x * x1.x + x1.y * x1.y + x1.z * x1.z + x1.w * x1.w;
    float dx = x0.x - x1.x; fp = fmaxf(fp, dx * dx);
    float dy = x0.y - x1.y; fp = fmaxf(fp, dy * dy);
    float dz = x0.z - x1.z; fp = fmaxf(fp, dz * dz);
    float dw = x0.w - x1.w; fp = fmaxf(fp, dw * dw);
  }
  na[i] = s0;
  nb[i] = s1;
  float px = xyz[3 * (size_t)i0 + 0];
  float py = xyz[3 * (size_t)i0 + 1];
  float pz = xyz[3 * (size_t)i0 + 2];
  sx[4 * (size_t)i + 0] = px;
  sx[4 * (size_t)i + 1] = py;
  sx[4 * (size_t)i + 2] = pz;
  sx[4 * (size_t)i + 3] = 0.f;     // pad so one K=4 WMMA does the 3-D dot
  nx[i] = px * px + py * py + pz * pz;
  float rr = fmaxf(fp - POS_THRESH, 0.f);
  posr[i] = rr * rr;
}

// ---------------------------------------------------------------------------
// Phase 2: one block per 16-row tile; 8 waves split the 16-column tiles.
// d2 = na[i] + nb[j] - 2*(A·B^T) via V_WMMA_F32_16X16X4_F32 (8x feat, 1x xyz).
// ---------------------------------------------------------------------------
__global__ __launch_bounds__(256) void mintile_kernel(
    const float* __restrict__ pf0, const float* __restrict__ pf1,
    const float* __restrict__ sx, const float* __restrict__ na,
    const float* __restrict__ nb, const float* __restrict__ nx,
    float* __restrict__ mn, int P) {
  const int tid  = threadIdx.x;
  const int wave = tid >> 5;
  const int lane = tid & 31;
  const int hv   = lane >> 4;      // 0: K=0,1 half; 1: K=2,3 half (A/B layout)
  const int r16  = lane & 15;
  const int rt   = blockIdx.x;     // row tile
  const int row  = rt * 16 + r16;

  // A fragments: lane (hv,r16) holds A[row][4s+2*hv .. +1]
  v2f a[8];
#pragma unroll
  for (int s = 0; s < 8; ++s) {
    float2 t = *reinterpret_cast<const float2*>(pf0 + (size_t)row * 32 + 4 * s + 2 * hv);
    a[s].x = t.x; a[s].y = t.y;
  }
  v2f axz;
  {
    float2 t = *reinterpret_cast<const float2*>(sx + (size_t)row * 4 + 2 * hv);
    axz.x = t.x; axz.y = t.y;
  }

  // C-layout row constants: VGPR r holds rows r (lanes 0-15) and r+8 (16-31)
  float na_r[8], nx_r[8];
#pragma unroll
  for (int r = 0; r < 8; ++r) {
    int m = rt * 16 + r + 8 * hv;
    na_r[r] = na[m];
    nx_r[r] = nx[m];
  }

  float rmin[8];
#pragma unroll
  for (int r = 0; r < 8; ++r) rmin[r] = BIG_VAL;

  const int ntiles = P >> 4;
  for (int j = wave; j < ntiles; j += WPB) {
    const int col = j * 16 + r16;
    // prefetch next column tile for this wave (global_prefetch_b8)
    if (col + WPB * 16 < P)
      __builtin_prefetch(pf1 + (size_t)(col + WPB * 16) * 32, 0, 1);

    v8f facc = {};
#pragma unroll
    for (int s = 0; s < 8; ++s) {
      float2 t = *reinterpret_cast<const float2*>(pf1 + (size_t)col * 32 + 4 * s + 2 * hv);
      v2f b; b.x = t.x; b.y = t.y;
      facc = wmma_f32_16x16x4(a[s], b, facc);
    }
    v8f xacc = {};
    {
      float2 t = *reinterpret_cast<const float2*>(sx + (size_t)col * 4 + 2 * hv);
      v2f b; b.x = t.x; b.y = t.y;
      xacc = wmma_f32_16x16x4(axz, b, xacc);
    }
    float nbn = nb[j * 16 + r16];
    float nxn = nx[j * 16 + r16];
#pragma unroll
    for (int r = 0; r < 8; ++r) {
      float d2  = fmaxf(na_r[r] + nbn - 2.f * facc[r], 0.f);
      float xd2 = fmaxf(nx_r[r] + nxn - 2.f * xacc[r], 0.f);
      // false_negative: sqrt(xd2 + 1e-7) > 0.15  <=>  xd2 + 1e-7 > 0.0225
      float cand = (xd2 + 1e-7f > MIN_DIST2) ? d2 : BIG_VAL;
      rmin[r] = fminf(rmin[r], cand);
    }
  }

  // min across the 16 lanes of each half (xor stays within half for off<16)
#pragma unroll
  for (int off = 1; off < 16; off <<= 1) {
#pragma unroll
    for (int r = 0; r < 8; ++r)
      rmin[r] = fminf(rmin[r], __shfl_xor(rmin[r], off, 32));
  }

  __shared__ float sm[16][WPB];
  if (r16 == 0) {
#pragma unroll
    for (int r = 0; r < 8; ++r) sm[r + 8 * hv][wave] = rmin[r];
  }
  __syncthreads();
  if (tid < 16) {
    float v = sm[tid][0];
#pragma unroll
    for (int w = 1; w < WPB; ++w) v = fminf(v, sm[tid][w]);
    mn[rt * 16 + tid] = v;
  }
}

// ---------------------------------------------------------------------------
// Phase 3: deterministic fixed-order reduction to the scalar loss.
// ---------------------------------------------------------------------------
__global__ __launch_bounds__(256) void reduce_kernel(
    const float* __restrict__ posr, const float* __restrict__ mn,
    float* __restrict__ out, int P) {
  __shared__ float sp[256];
  __shared__ float sn[256];
  int t = threadIdx.x;
  float ap = 0.f, an = 0.f;
  for (int i = t; i < P; i += 256) {
    ap += posr[i];
    float r = fmaxf(NEG_THRESH - mn[i], 0.f);
    an += r * r;
  }
  sp[t] = ap;
  sn[t] = an;
  __syncthreads();
  for (int s = 128; s > 0; s >>= 1) {
    if (t < s) { sp[t] += sp[t + s]; sn[t] += sn[t + s]; }
    __syncthreads();
  }
  if (t == 0) out[0] = (sp[0] + sn[0]) / (float)P;
}

extern "C" void kernel_launch(void* const* d_in, const int* in_sizes, int n_in,
                              void* d_out, int out_size, void* d_ws, size_t ws_size,
                              hipStream_t stream) {
  (void)n_in; (void)out_size; (void)ws_size;
  const float* F0   = (const float*)d_in[0];
  const float* F1   = (const float*)d_in[1];
  const int*   prs  = (const int*)d_in[2];
  const float* xyz  = (const float*)d_in[3];
  float*       out  = (float*)d_out;
  const int P = in_sizes[2] / 2;   // 8192

  float* ws   = (float*)d_ws;
  float* pf0  = ws;                          // P*32
  float* pf1  = pf0 + (size_t)P * 32;        // P*32
  float* sx   = pf1 + (size_t)P * 32;        // P*4 (xyz padded)
  float* na   = sx  + (size_t)P * 4;         // P
  float* nb   = na  + P;                     // P
  float* nx   = nb  + P;                     // P
  float* posr = nx  + P;                     // P
  float* mn   = posr + P;                    // P

  gather_kernel<<<(P + 255) / 256, 256, 0, stream>>>(F0, F1, prs, xyz,
                                                     pf0, pf1, sx, na, nb, nx, posr, P);
  mintile_kernel<<<P / 16, 256, 0, stream>>>(pf0, pf1, sx, na, nb, nx, mn, P);
  reduce_kernel<<<1, 256, 0, stream>>>(posr, mn, out, P);
}